// ShelvingEQ_56538949485207
// MI455X (gfx1250) — compile-verified
//
#include <hip/hip_runtime.h>
#include <math.h>

typedef __attribute__((ext_vector_type(2))) float v2f;
typedef __attribute__((ext_vector_type(8))) float v8f;

#define SR   44100.0f
#define NB   32          // batch (channels)
#define NS   262144      // samples per channel
#define LCH  256         // chunk length
#define CCH  (NS / LCH)  // 1024 chunks per channel
#define CPG  64          // chunks per workgroup in K3
#define TSTR (CPG + 1)   // time-major LDS tile row stride (65): conflict-free both phases

// workspace layout in floats
#define WS_SOS 0                         // 32 * 20   (b0,b1,b2,a1,a2 x 4 sections)
#define WS_H   (WS_SOS + NB * 20)        // 32 * 256 * 8  initial-state impulse response
#define WS_M   (WS_H + NB * LCH * 8)     // 32 * 8 * 8    chunk transition matrix
#define WS_F   (WS_M + NB * 64)          // 32 * 1024 * 8 zero-state chunk-final states
#define WS_S0  (WS_F + NB * CCH * 8)     // 32 * 1024 * 8 chunk initial states

// ---------------------------------------------------------------- coefficients
__device__ inline void bq_coeffs(int type, float gain_db, float f0, float q,
                                 float* __restrict__ out5) {
  float A  = __powf(10.0f, gain_db * 0.025f);   // 10^(g/40)
  float w0 = 2.0f * 3.14159265358979323f * (f0 / SR);
  float sw, cw;
  __sincosf(w0, &sw, &cw);
  float alpha = sw / (2.0f * q);
  float sA = sqrtf(A);
  float b0, b1, b2, a0, a1, a2;
  if (type == 0) {            // highpass
    b0 = (1.0f + cw) * 0.5f; b1 = -(1.0f + cw); b2 = b0;
    a0 = 1.0f + alpha; a1 = -2.0f * cw; a2 = 1.0f - alpha;
  } else if (type == 1) {     // lowpass
    b0 = (1.0f - cw) * 0.5f; b1 = 1.0f - cw; b2 = b0;
    a0 = 1.0f + alpha; a1 = -2.0f * cw; a2 = 1.0f - alpha;
  } else if (type == 2) {     // low shelf
    b0 = A * ((A + 1.0f) - (A - 1.0f) * cw + 2.0f * sA * alpha);
    b1 = 2.0f * A * ((A - 1.0f) - (A + 1.0f) * cw);
    b2 = A * ((A + 1.0f) - (A - 1.0f) * cw - 2.0f * sA * alpha);
    a0 = (A + 1.0f) + (A - 1.0f) * cw + 2.0f * sA * alpha;
    a1 = -2.0f * ((A - 1.0f) + (A + 1.0f) * cw);
    a2 = (A + 1.0f) - (A - 1.0f) * cw - 2.0f * sA * alpha;
  } else {                    // high shelf
    b0 = A * ((A + 1.0f) + (A - 1.0f) * cw + 2.0f * sA * alpha);
    b1 = -2.0f * A * ((A - 1.0f) + (A + 1.0f) * cw);
    b2 = A * ((A + 1.0f) + (A - 1.0f) * cw - 2.0f * sA * alpha);
    a0 = (A + 1.0f) - (A - 1.0f) * cw + 2.0f * sA * alpha;
    a1 = 2.0f * ((A - 1.0f) - (A + 1.0f) * cw);
    a2 = (A + 1.0f) + (A - 1.0f) * cw - 2.0f * sA * alpha;
  }
  float inv = 1.0f / a0;
  out5[0] = b0 * inv; out5[1] = b1 * inv; out5[2] = b2 * inv;
  out5[3] = a1 * inv; out5[4] = a2 * inv;
}

__global__ void k1_sos(const float* __restrict__ cp, float* __restrict__ ws) {
  int b = threadIdx.x;
  if (b >= NB) return;
  const float pmin[10] = {20.0f, 0.1f, -12.0f, 20.0f, 0.1f, -12.0f, 4000.0f, 0.1f, 4000.0f, 0.1f};
  const float pmax[10] = {2000.0f, 10.0f, 12.0f, 4000.0f, 10.0f, 12.0f, 24000.0f, 10.0f, 24000.0f, 10.0f};
  float p[10];
#pragma unroll
  for (int i = 0; i < 10; ++i)
    p[i] = fmaf(cp[b * 10 + i], pmax[i] - pmin[i], pmin[i]);
  float* s = ws + WS_SOS + b * 20;
  bq_coeffs(0, 0.0f, p[0], p[1], s + 0);    // highpass
  bq_coeffs(2, p[2], p[3], p[4], s + 5);    // low shelf
  bq_coeffs(3, p[5], p[6], p[7], s + 10);   // high shelf
  bq_coeffs(1, 0.0f, p[8], p[9], s + 15);   // lowpass
}

// cascade single step (DF2T), x in, y out, st[8] updated
__device__ inline float cascade_step(const float* __restrict__ cf, float* __restrict__ st, float x) {
#pragma unroll
  for (int k = 0; k < 4; ++k) {
    float b0 = cf[k * 5], b1 = cf[k * 5 + 1], b2 = cf[k * 5 + 2];
    float a1 = cf[k * 5 + 3], a2 = cf[k * 5 + 4];
    float yv = fmaf(b0, x, st[k * 2]);
    st[k * 2]     = fmaf(b1, x, fmaf(-a1, yv, st[k * 2 + 1]));
    st[k * 2 + 1] = fmaf(b2, x, -a2 * yv);
    x = yv;
  }
  return x;
}

// ----------------------------------------------- K2: impulse response + transition
__global__ void k2_impulse(float* __restrict__ ws) {
  int tid = threadIdx.x;          // 0..255
  int b = tid >> 3, j = tid & 7;
  float cf[20];
#pragma unroll
  for (int i = 0; i < 20; ++i) cf[i] = ws[WS_SOS + b * 20 + i];
  float st[8];
#pragma unroll
  for (int i = 0; i < 8; ++i) st[i] = 0.0f;
  st[j] = 1.0f;
  float* Hb = ws + WS_H + (size_t)b * LCH * 8;
  for (int t = 0; t < LCH; ++t)
    Hb[t * 8 + j] = cascade_step(cf, st, 0.0f);
  float* M = ws + WS_M + b * 64;          // M[row][col=j] = final state component
#pragma unroll
  for (int k = 0; k < 8; ++k) M[k * 8 + j] = st[k];
}

// ----------------------------------------------- K3: zero-state chunk filtering
// 64 threads / 64 chunks of one channel. Time-major LDS tile [LCH][65]:
//   async-LDS write bank = (tid + it/4) % 64, compute read bank = (t + tid) % 64.
// Global <-> LDS moved with CDNA5 async ops (ASYNCcnt), no VGPR round-trip.
__global__ void __launch_bounds__(CPG) k3_zerostate(const float* __restrict__ xin,
                                                    float* __restrict__ yout,
                                                    float* __restrict__ ws) {
  __shared__ float tile[LCH * TSTR];
  int tid = threadIdx.x;                 // 0..63
  int cg  = blockIdx.x;                  // chunk group 0..15
  int b   = blockIdx.y;                  // channel 0..31
  size_t gbase = (size_t)b * NS + (size_t)cg * CPG * LCH;
  unsigned lbase = (unsigned)(uintptr_t)tile;  // generic shared ptr low 32 bits = LDS offset

  // async global -> LDS : element e = it*64+tid, chunk c = it/4, time t = (it&3)*64+tid
  for (int it = 0; it < 4 * CPG; ++it) {
    int t = ((it & 3) << 6) + tid;
    int c = it >> 2;
    unsigned long long g = (unsigned long long)(uintptr_t)(xin + gbase + (size_t)(it << 6) + tid);
    unsigned l = lbase + (unsigned)((t * TSTR + c) << 2);
    asm volatile("global_load_async_to_lds_b32 %0, %1, off" :: "v"(l), "v"(g) : "memory");
  }
  asm volatile("s_wait_asynccnt 0x0" ::: "memory");
  __syncthreads();

  float cf[20];
#pragma unroll
  for (int i = 0; i < 20; ++i) cf[i] = ws[WS_SOS + b * 20 + i];
  float st[8];
#pragma unroll
  for (int i = 0; i < 8; ++i) st[i] = 0.0f;

  for (int t = 0; t < LCH; ++t) {
    float* p = tile + t * TSTR + tid;
    *p = cascade_step(cf, st, *p);
  }

  int cidx = cg * CPG + tid;
  float* f = ws + WS_F + ((size_t)b * CCH + cidx) * 8;
#pragma unroll
  for (int k = 0; k < 8; ++k) f[k] = st[k];
  __syncthreads();

  // async LDS -> global (mirror of load)
  for (int it = 0; it < 4 * CPG; ++it) {
    int t = ((it & 3) << 6) + tid;
    int c = it >> 2;
    unsigned long long g = (unsigned long long)(uintptr_t)(yout + gbase + (size_t)(it << 6) + tid);
    unsigned l = lbase + (unsigned)((t * TSTR + c) << 2);
    asm volatile("global_store_async_from_lds_b32 %0, %1, off" :: "v"(g), "v"(l) : "memory");
  }
  asm volatile("s_wait_asynccnt 0x0" ::: "memory");
}

// ----------------------------------------------- K4: two-level chunk-state scan
// 256 threads per channel = 32 teams of 8 lanes. M is time-invariant, so:
//  A) team g: F_g = sum_i M^(31-i) f[g*32+i]      (32 parallel, depth 32)
//  B) lanes 0..7: M32 = M^32 by 5 squarings        (depth ~40 shuffled FMAs)
//  C) lanes 0..7: scan 32 group states with M32    (depth 32)
//  D) team g: expand its 32 chunk states           (32 parallel, depth 32)
__global__ void __launch_bounds__(256) k4_scan(float* __restrict__ ws) {
  __shared__ float Fall[32][8];
  __shared__ float SG[32][8];
  int b   = blockIdx.x;
  int tid = threadIdx.x;
  int g   = tid >> 3;                    // team 0..31
  int j   = tid & 7;                     // state component
  int lsub = (tid & 31) & ~7;            // team base lane within wave

  float mrow[8];
#pragma unroll
  for (int k = 0; k < 8; ++k) mrow[k] = ws[WS_M + b * 64 + j * 8 + k];
  const float* f = ws + WS_F + (size_t)b * CCH * 8 + (size_t)g * 32 * 8;

  // Phase A
  float F = 0.0f;
  for (int i = 0; i < 32; ++i) {
    float sv[8];
#pragma unroll
    for (int k = 0; k < 8; ++k) sv[k] = __shfl(F, lsub + k, 32);
    float acc = f[i * 8 + j];
#pragma unroll
    for (int k = 0; k < 8; ++k) acc = fmaf(mrow[k], sv[k], acc);
    F = acc;
  }
  Fall[g][j] = F;
  __syncthreads();

  // Phases B + C on lanes 0..7 of wave 0
  if (tid < 8) {
    float r[8];
#pragma unroll
    for (int k = 0; k < 8; ++k) r[k] = mrow[k];
    for (int sq = 0; sq < 5; ++sq) {     // r <- r * r  (row j of P^2)
      float nr[8];
#pragma unroll
      for (int k = 0; k < 8; ++k) {
        float acc = 0.0f;
#pragma unroll
        for (int l = 0; l < 8; ++l) acc = fmaf(r[l], __shfl(r[k], l, 32), acc);
        nr[k] = acc;
      }
#pragma unroll
      for (int k = 0; k < 8; ++k) r[k] = nr[k];
    }
    float s = 0.0f;                       // group-boundary scan with M^32
    for (int gg = 0; gg < 32; ++gg) {
      SG[gg][j] = s;
      float sv[8];
#pragma unroll
      for (int k = 0; k < 8; ++k) sv[k] = __shfl(s, k, 32);
      float acc = Fall[gg][j];
#pragma unroll
      for (int k = 0; k < 8; ++k) acc = fmaf(r[k], sv[k], acc);
      s = acc;
    }
  }
  __syncthreads();

  // Phase D
  float s = SG[g][j];
  float* s0 = ws + WS_S0 + (size_t)b * CCH * 8 + (size_t)g * 32 * 8;
  for (int i = 0; i < 32; ++i) {
    s0[i * 8 + j] = s;
    float sv[8];
#pragma unroll
    for (int k = 0; k < 8; ++k) sv[k] = __shfl(s, lsub + k, 32);
    float acc = f[i * 8 + j];
#pragma unroll
    for (int k = 0; k < 8; ++k) acc = fmaf(mrow[k], sv[k], acc);
    s = acc;
  }
}

// ----------------------------------------------- K5: WMMA state correction GEMM
// y[c, t] += H[t, 0:8] . s0[c, 0:8]  as D(16x16) = A(16x4) B(4x16) + C, K 8 = 4+4
__global__ void __launch_bounds__(32) k5_correct(float* __restrict__ y,
                                                 const float* __restrict__ ws) {
  int b    = blockIdx.y;
  int c0   = blockIdx.x * 16;            // 16-chunk tile
  int lane = threadIdx.x;
  int half = lane >> 4;
  int l    = lane & 15;
  int k0   = half * 2;

  // A: S0 tile (16 chunks x 4 states). lanes 0-15: K0/K1, lanes 16-31: K2/K3
  const float* s0 = ws + WS_S0 + ((size_t)b * CCH + c0 + l) * 8;
  v2f a03, a47;
  a03.x = s0[k0];     a03.y = s0[k0 + 1];
  a47.x = s0[4 + k0]; a47.y = s0[5 + k0];

  const float* Hb = ws + WS_H + (size_t)b * LCH * 8;
  float* yb = y + (size_t)b * NS + (size_t)c0 * LCH;

  for (int t0 = 0; t0 < LCH; t0 += 16) {
    const float* h = Hb + (size_t)(t0 + l) * 8;    // B: H^T tile (4 states x 16 times)
    v2f b03, b47;
    b03.x = h[k0];     b03.y = h[k0 + 1];
    b47.x = h[4 + k0]; b47.y = h[5 + k0];

    v8f acc = {};
    acc = __builtin_amdgcn_wmma_f32_16x16x4_f32(false, a03, false, b03,
                                                (short)0, acc, false, false);
    acc = __builtin_amdgcn_wmma_f32_16x16x4_f32(false, a47, false, b47,
                                                (short)0, acc, false, false);
#pragma unroll
    for (int j = 0; j < 8; ++j) {        // D: VGPR j -> M = j + 8*half, N = lane&15
      int m = j + 8 * half;
      size_t idx = (size_t)m * LCH + t0 + l;
      yb[idx] += acc[j];
    }
  }
}

extern "C" void kernel_launch(void* const* d_in, const int* in_sizes, int n_in,
                              void* d_out, int out_size, void* d_ws, size_t ws_size,
                              hipStream_t stream) {
  const float* x  = (const float*)d_in[0];   // (32, 1, 262144)
  const float* cp = (const float*)d_in[1];   // (32, 10, 1)
  float* y  = (float*)d_out;
  float* ws = (float*)d_ws;

  k1_sos<<<1, 32, 0, stream>>>(cp, ws);
  k2_impulse<<<1, 256, 0, stream>>>(ws);
  k3_zerostate<<<dim3(CCH / CPG, NB), CPG, 0, stream>>>(x, y, ws);
  k4_scan<<<NB, 256, 0, stream>>>(ws);
  k5_correct<<<dim3(CCH / 16, NB), 32, 0, stream>>>(y, ws);
}